// UndistortLayer_53936199303600
// MI455X (gfx1250) — compile-verified
//
#include <hip/hip_runtime.h>

// Undistort (radial lens model) + bilinear gather, C=3, H=W=2048, fp32.
// Memory-bound: ~100 MB traffic -> ~4.3 us at 23.3 TB/s HBM on MI455X.
// One thread = one (y,x) pixel, all 3 channels (reuses the coordinate map
// and the 4 gather index pairs; 12 independent global_load_b32 in flight).
//
// WMMA/TDM are deliberately NOT used: every pixel has data-dependent gather
// addresses and per-pixel blend weights, so there is no wave-shared matrix
// operand (WMMA) and no dense tile for the tensor data mover. Per-lane
// global_load gathers are the correct CDNA5 path for this op.

#define CC 3
#define HH 2048
#define WW 2048
#define LOG2W 11

__global__ __launch_bounds__(256) void undistort_kernel(
    const float* __restrict__ im,
    const float* __restrict__ kp,
    const float* __restrict__ dxp,
    const float* __restrict__ dyp,
    float* __restrict__ out)
{
    // Uniform scalars (promoted to s_load: uniform, readonly).
    const float k  = kp[0];
    const float dx = dxp[0];
    const float dy = dyp[0];

    const int idx = blockIdx.x * 256 + threadIdx.x;   // 0 .. W*H-1
    const int x = idx & (WW - 1);
    const int y = idx >> LOG2W;

    // Coordinate map. /w and /h are exact multiplies (powers of two).
    const float invw = 1.0f / (float)WW;
    const float invh = 1.0f / (float)HH;
    float xur = ((float)x - dx) * invw - 0.5f;
    float yur = ((float)y - dy) * invh - 0.5f;

    // rd*cos(theta) = xur / (1 - k*ru^2); rd*sin(theta) = yur / (1 - k*ru^2).
    // (At ru==0 this matches the reference's atan2 path exactly.)
    float r2 = fmaf(xur, xur, yur * yur);
    float s  = 1.0f / fmaf(-k, r2, 1.0f);       // full-precision fp32 divide
    float xd = fmaf(xur * s + 0.5f, (float)WW, dx);
    float yd = fmaf(yur * s + 0.5f, (float)HH, dy);

    float xf = floorf(xd);
    float yf = floorf(yd);
    float xc = ceilf(xd);
    float yc = ceilf(yd);

    float ox  = xd - xf;
    float oy  = yd - yf;
    float onx = 1.0f - ox;
    float ony = 1.0f - oy;

    // valid -> keep weights; invalid -> all-zero weights. Gathered values are
    // finite, so 0*v == 0 reproduces where(valid, blend, 0) exactly.
    const bool valid = (xf >= 0.0f) && (xc < (float)WW) &&
                       (yf >= 0.0f) && (yc < (float)HH);
    const float vm = valid ? 1.0f : 0.0f;

    float w00 = (onx * ony) * vm;
    float w01 = (ox  * ony) * vm;
    float w10 = (onx * oy ) * vm;
    float w11 = (ox  * oy ) * vm;

    // Clip indices exactly like the reference, then gather unconditionally.
    int xfi = min(max((int)xf, 0), WW - 1);
    int xci = min(max((int)xc, 0), WW - 1);
    int yfi = min(max((int)yf, 0), HH - 1);
    int yci = min(max((int)yc, 0), HH - 1);

    const int rowf = yfi << LOG2W;
    const int rowc = yci << LOG2W;
    const int i00 = rowf + xfi;
    const int i01 = rowf + xci;
    const int i10 = rowc + xfi;
    const int i11 = rowc + xci;

    const int chs = WW * HH;   // channel stride (elements)

    const float* __restrict__ p0 = im;
    const float* __restrict__ p1 = im + chs;
    const float* __restrict__ p2 = im + 2 * chs;

    // 12 independent gathers; __restrict__ lets the scheduler batch them
    // ahead of a single s_wait_loadcnt.
    float a00 = p0[i00], a01 = p0[i01], a10 = p0[i10], a11 = p0[i11];
    float b00 = p1[i00], b01 = p1[i01], b10 = p1[i10], b11 = p1[i11];
    float c00 = p2[i00], c01 = p2[i01], c10 = p2[i10], c11 = p2[i11];

    float r0 = fmaf(w00, a00, fmaf(w01, a01, fmaf(w10, a10, w11 * a11)));
    float r1 = fmaf(w00, b00, fmaf(w01, b01, fmaf(w10, b10, w11 * b11)));
    float r2c = fmaf(w00, c00, fmaf(w01, c01, fmaf(w10, c10, w11 * c11)));

    out[idx]           = r0;
    out[idx + chs]     = r1;
    out[idx + 2 * chs] = r2c;
}

extern "C" void kernel_launch(void* const* d_in, const int* in_sizes, int n_in,
                              void* d_out, int out_size, void* d_ws, size_t ws_size,
                              hipStream_t stream) {
    const float* im = (const float*)d_in[0];
    const float* k  = (const float*)d_in[1];
    const float* dx = (const float*)d_in[2];
    const float* dy = (const float*)d_in[3];
    float* out = (float*)d_out;

    const int npix = WW * HH;             // 4,194,304 pixels
    dim3 grid(npix / 256), block(256);    // 16384 blocks x 8 wave32
    undistort_kernel<<<grid, block, 0, stream>>>(im, k, dx, dy, out);
}